// YOLOLoss_44074954391819
// MI455X (gfx1250) — compile-verified
//
#include <hip/hip_runtime.h>
#include <cstdint>

// Problem constants (from setup_inputs): B=8, N=32768, M=100, D=85
#define B_ 8
#define N_ 32768
#define M_ 100
#define D_ 85
#define S_ 128                 // N-slices per batch (kernel-1 blocks per batch)
#define CHUNK (N_ / S_)        // 256 pred rows per block
#define TILE 128               // rows staged per async tile
#define NTILES (CHUNK / TILE)  // 2 tiles, double buffered

// ---- CDNA5 async LDS staging -------------------------------------------------
// Loads 16 bytes (one box: x1,y1,x2,y2) from a 4B-aligned global address into
// LDS using the gfx1250 async-to-LDS path (ASYNCcnt). The instruction's
// immediate offset is applied to BOTH the LDS and global addresses (ISA
// CDNA5 ISA), so one LDS base + one 64b address VGPR pair works
// for all four dwords.
__device__ __forceinline__ void async_ld_box16(unsigned lds_addr,
                                               unsigned long long gaddr) {
  asm volatile(
      "global_load_async_to_lds_b32 %0, %1, off\n\t"
      "global_load_async_to_lds_b32 %0, %1, off offset:4\n\t"
      "global_load_async_to_lds_b32 %0, %1, off offset:8\n\t"
      "global_load_async_to_lds_b32 %0, %1, off offset:12"
      :: "v"(lds_addr), "v"(gaddr)
      : "memory");
}

__device__ __forceinline__ void wait_async0() {
#if __has_builtin(__builtin_amdgcn_s_wait_asynccnt)
  __builtin_amdgcn_s_wait_asynccnt(0);
#else
  asm volatile("s_wait_asynccnt 0" ::: "memory");
#endif
}

// ---- Kernel 1: per-slice GIoU max/argmax partials ---------------------------
// grid = (S_, B_), block = 128 threads (4 wave32). Lane t owns target t.
__global__ __launch_bounds__(TILE) void k_giou_partial(
    const float* __restrict__ pred,   // [B, N, 85]
    const float* __restrict__ tgt,    // [B, M, 85]
    float* __restrict__ pmax,         // [B, S_, M]
    int*   __restrict__ pidx)         // [B, S_, M]
{
  __shared__ float4 sT[TILE];         // target boxes (first 100 valid)
  __shared__ float4 sP[2][TILE];      // double-buffered pred box tiles

  const int t = threadIdx.x;
  const int s = blockIdx.x;
  const int b = blockIdx.y;
  const int base = s * CHUNK;

  // Stage target boxes for this batch (lanes 0..99), async.
  if (t < M_) {
    async_ld_box16((unsigned)(uintptr_t)&sT[t],
                   (unsigned long long)(uintptr_t)(tgt + ((size_t)b * M_ + t) * D_));
  }
  // Stage pred tile 0, async.
  async_ld_box16((unsigned)(uintptr_t)&sP[0][t],
                 (unsigned long long)(uintptr_t)(pred + ((size_t)b * N_ + base + t) * D_));
  wait_async0();
  __syncthreads();

  const int tt = (t < M_) ? t : 0;    // lanes >=100 shadow target 0 (results discarded)
  const float4 tb = sT[tt];
  const float area_t = (tb.z - tb.x) * (tb.w - tb.y);

  float best  = -INFINITY;
  int   bestn = 0;

  for (int k = 0; k < NTILES; ++k) {
    if (k + 1 < NTILES) {
      // Prefetch next tile into the other buffer while computing this one.
      async_ld_box16((unsigned)(uintptr_t)&sP[(k + 1) & 1][t],
                     (unsigned long long)(uintptr_t)
                         (pred + ((size_t)b * N_ + base + (k + 1) * TILE + t) * D_));
    }
    const float4* tp = sP[k & 1];
    const int n0 = base + k * TILE;
#pragma unroll 4
    for (int r = 0; r < TILE; ++r) {
      const float4 p = tp[r];                       // broadcast ds_load_b128
      const float area_p = (p.z - p.x) * (p.w - p.y);
      // intersection
      const float ltx = fmaxf(p.x, tb.x), lty = fmaxf(p.y, tb.y);
      const float rbx = fminf(p.z, tb.z), rby = fminf(p.w, tb.w);
      const float w = fmaxf(rbx - ltx, 0.0f), h = fmaxf(rby - lty, 0.0f);
      const float inter = w * h;
      const float uni = area_p + area_t - inter;
      const float iou = inter / uni;
      // enclosing box
      const float lex = fminf(p.x, tb.x), ley = fminf(p.y, tb.y);
      const float rex = fmaxf(p.z, tb.z), rey = fmaxf(p.w, tb.w);
      const float we = fmaxf(rex - lex, 0.0f), he = fmaxf(rey - ley, 0.0f);
      const float ae = we * he;
      const float g = iou - (ae - uni) / ae;
      const int n = n0 + r;
      if (g > best) { best = g; bestn = n; }        // strict > keeps first occurrence
    }
    wait_async0();
    __syncthreads();
  }

  if (t < M_) {
    pmax[((size_t)b * S_ + s) * M_ + t] = best;
    pidx[((size_t)b * S_ + s) * M_ + t] = bestn;
  }
}

// ---- Kernel 2: final reduce per (b,m) + gather winner rows ------------------
// grid = B*M blocks, block = 32 (one wave32).
__global__ __launch_bounds__(32) void k_reduce_gather(
    const float* __restrict__ pred,
    const float* __restrict__ pmax,
    const int*   __restrict__ pidx,
    float* __restrict__ out)          // [B] loss | [B,M] max | [B,M,85] objects
{
  const int bm = blockIdx.x;          // 0 .. B*M-1
  const int b = bm / M_;
  const int m = bm % M_;
  const int l = threadIdx.x;

  float v = -INFINITY;
  int   i = 0;
#pragma unroll
  for (int q = 0; q < S_ / 32; ++q) {
    const int s = l + q * 32;
    const float v2 = pmax[((size_t)b * S_ + s) * M_ + m];
    const int   i2 = pidx[((size_t)b * S_ + s) * M_ + m];
    if (v2 > v || (v2 == v && i2 < i)) { v = v2; i = i2; }
  }
  // Butterfly: lexicographic (value asc, index desc) max is associative &
  // commutative, so every lane converges to the same (v, i).
#pragma unroll
  for (int off = 16; off; off >>= 1) {
    const float vo = __shfl_xor(v, off, 32);
    const int   io = __shfl_xor(i, off, 32);
    if (vo > v || (vo == v && io < i)) { v = vo; i = io; }
  }

  if (l == 0) out[B_ + bm] = v;       // max_gious

  const float* row = pred + ((size_t)b * N_ + i) * D_;
  float* orow = out + B_ + (size_t)B_ * M_ + (size_t)bm * D_;
  for (int j = l; j < D_; j += 32) orow[j] = row[j];   // objects
}

// ---- Kernel 3: masked coord loss per batch ----------------------------------
// grid = B blocks, block = 128.
__global__ __launch_bounds__(128) void k_coord_loss(
    const int* __restrict__ num_targets,
    float* __restrict__ out)
{
  __shared__ float wsum[4];
  const int b = blockIdx.x;
  const int t = threadIdx.x;

  float v = 0.0f;
  if (t < M_) {
    const float mg = out[B_ + b * M_ + t];
    const float mask = (t < num_targets[b]) ? 1.0f : 0.0f;
    v = (1.0f - mg) * mask;
  }
#pragma unroll
  for (int off = 16; off; off >>= 1) v += __shfl_xor(v, off, 32);
  if ((t & 31) == 0) wsum[t >> 5] = v;
  __syncthreads();
  if (t == 0) out[b] = wsum[0] + wsum[1] + wsum[2] + wsum[3];
}

extern "C" void kernel_launch(void* const* d_in, const int* in_sizes, int n_in,
                              void* d_out, int out_size, void* d_ws, size_t ws_size,
                              hipStream_t stream) {
  const float* pred = (const float*)d_in[0];   // [8,32768,85] f32
  const float* tgt  = (const float*)d_in[1];   // [8,100,85]   f32
  const int*   ntgt = (const int*)  d_in[2];   // [8]          i32
  float* out = (float*)d_out;                  // 8 + 800 + 68000 f32

  float* pmax = (float*)d_ws;                                        // B*S*M f32
  int*   pidx = (int*)((char*)d_ws + sizeof(float) * B_ * S_ * M_);  // B*S*M i32

  k_giou_partial<<<dim3(S_, B_), TILE, 0, stream>>>(pred, tgt, pmax, pidx);
  k_reduce_gather<<<B_ * M_, 32, 0, stream>>>(pred, pmax, pidx, out);
  k_coord_loss<<<B_, 128, 0, stream>>>(ntgt, out);
}